// MipCubemapEncoder_85074712199702
// MI455X (gfx1250) — compile-verified
//
#include <hip/hip_runtime.h>

// ---------------- configuration ----------------
#define THREADS 256
#define PPT 8                        // points per thread
#define PPB (THREADS * PPT)          // points per block = 2048

#define CCH 6                        // channels per level
#define LV0 4
#define LV1 16
#define LV2 64
#define LV3 256

#define N0 (6 * LV0 * LV0 * CCH)     // 576 floats
#define N1 (6 * LV1 * LV1 * CCH)     // 9216 floats
#define N2 (6 * LV2 * LV2 * CCH)     // 147456 floats
#define N3 (6 * LV3 * LV3 * CCH)     // 2359296 floats
#define LDS_FLOATS (N0 + N1)         // 9792 floats = 39168 bytes

#define WS_OFF0 0
#define WS_OFF1 (N0)
#define WS_OFF2 (N0 + N1)
#define WS_OFF3 (N0 + N1 + N2)
#define WS_FLOATS (N0 + N1 + N2 + N3)

typedef float vf4 __attribute__((ext_vector_type(4)));

// gfx1250 async global->LDS copy path (guarded; falls back to ds_store)
#if defined(__has_builtin)
#if __has_builtin(__builtin_amdgcn_global_load_async_to_lds_b128) && \
    __has_builtin(__builtin_amdgcn_s_wait_asynccnt)
#define USE_ASYNC_LDS 1
#endif
#endif
#ifndef USE_ASYNC_LDS
#define USE_ASYNC_LDS 0
#endif

#if USE_ASYNC_LDS
typedef int v4i_t __attribute__((ext_vector_type(4)));
typedef __attribute__((address_space(1))) v4i_t* global_v4i_p;
typedef __attribute__((address_space(3))) v4i_t* shared_v4i_p;
#endif

// ---------------- prologue: (6,C,L,L) -> (6,L,L,C) transpose ----------------
// Regular-temporal stores on purpose: the transposed tables should stay in L2
// for the gather kernel.
__global__ void transpose_emb_kernel(const float* __restrict__ src,
                                     float* __restrict__ dst, int L) {
    int total = 6 * L * L;           // texels
    int LL = L * L;
    for (int i = blockIdx.x * blockDim.x + threadIdx.x; i < total;
         i += gridDim.x * blockDim.x) {
        int x = i % L;
        int t = i / L;
        int y = t % L;
        int f = t / L;
        const float* s = src + (size_t)f * CCH * LL + y * L + x;
        float* d = dst + (size_t)i * CCH;
#pragma unroll
        for (int c = 0; c < CCH; ++c) d[c] = s[c * LL];
    }
}

// ---------------- gather helpers ----------------
__device__ __forceinline__ void bilinear_setup(float u, float w, int L,
                                               int& x0i, int& x1i, int& y0i, int& y1i,
                                               float& w00, float& w01, float& w10, float& w11) {
    float fx = u * (float)L - 0.5f;
    float fy = w * (float)L - 0.5f;
    float x0f = floorf(fx);
    float y0f = floorf(fy);
    float wx = fx - x0f;
    float wy = fy - y0f;
    int x0 = (int)x0f;
    int y0 = (int)y0f;
    x0i = min(max(x0, 0), L - 1);
    x1i = min(max(x0 + 1, 0), L - 1);
    y0i = min(max(y0, 0), L - 1);
    y1i = min(max(y0 + 1, 0), L - 1);
    float omx = 1.0f - wx, omy = 1.0f - wy;
    w00 = omx * omy;
    w01 = wx * omy;
    w10 = omx * wy;
    w11 = wx * wy;
}

// channel-last table: (6, L, L, C)
__device__ __forceinline__ void gather_chlast(const float* __restrict__ tbl,
                                              int face, float u, float w, int L,
                                              float* __restrict__ r) {
    int x0i, x1i, y0i, y1i;
    float w00, w01, w10, w11;
    bilinear_setup(u, w, L, x0i, x1i, y0i, y1i, w00, w01, w10, w11);
    int fb = face * L;
    const float* p00 = tbl + ((fb + y0i) * L + x0i) * CCH;
    const float* p01 = tbl + ((fb + y0i) * L + x1i) * CCH;
    const float* p10 = tbl + ((fb + y1i) * L + x0i) * CCH;
    const float* p11 = tbl + ((fb + y1i) * L + x1i) * CCH;
#pragma unroll
    for (int c = 0; c < CCH; ++c)
        r[c] = p00[c] * w00 + p01[c] * w01 + p10[c] * w10 + p11[c] * w11;
}

// channel-major table: (6, C, L, L)  (fallback when ws too small)
__device__ __forceinline__ void gather_chmajor(const float* __restrict__ tbl,
                                               int face, float u, float w, int L,
                                               float* __restrict__ r) {
    int x0i, x1i, y0i, y1i;
    float w00, w01, w10, w11;
    bilinear_setup(u, w, L, x0i, x1i, y0i, y1i, w00, w01, w10, w11);
    int LL = L * L;
    const float* base = tbl + (size_t)face * CCH * LL;
    int i00 = y0i * L + x0i;
    int i01 = y0i * L + x1i;
    int i10 = y1i * L + x0i;
    int i11 = y1i * L + x1i;
#pragma unroll
    for (int c = 0; c < CCH; ++c) {
        const float* pc = base + c * LL;
        r[c] = pc[i00] * w00 + pc[i01] * w01 + pc[i10] * w10 + pc[i11] * w11;
    }
}

// stage a level into LDS in channel-last order directly from (6,C,L,L)
__device__ __forceinline__ void stage_level_transpose(float* __restrict__ dst,
                                                      const float* __restrict__ src,
                                                      int L) {
    int total = 6 * L * L * CCH;
    int LL = L * L;
    for (int i = threadIdx.x; i < total; i += THREADS) {
        int c = i % CCH;
        int t = i / CCH;
        int x = t % L;
        t /= L;
        int y = t % L;
        int f = t / L;
        dst[i] = src[(f * CCH + c) * LL + y * L + x];
    }
}

// ---------------- main kernel ----------------
template <bool USE_WS>
__global__ void __launch_bounds__(THREADS)
cubemap_encode_kernel(const float* __restrict__ pts,
                      const float* __restrict__ emb0,
                      const float* __restrict__ emb1,
                      const float* __restrict__ tab2,   // ws-transposed or raw emb2
                      const float* __restrict__ tab3,   // ws-transposed or raw emb3
                      const float* __restrict__ ws01,   // transposed lvl0+lvl1 (USE_WS)
                      const float* __restrict__ fail,
                      float* __restrict__ out, int n) {
    __shared__ float lds[LDS_FLOATS];

    if (USE_WS) {
        // levels 0+1 are already channel-last in ws: straight wide copy
        const vf4* src = (const vf4*)ws01;
        vf4* dst = (vf4*)lds;
#if USE_ASYNC_LDS
        for (int i = threadIdx.x; i < LDS_FLOATS / 4; i += THREADS) {
            __builtin_amdgcn_global_load_async_to_lds_b128(
                (global_v4i_p)(src + i),
                (shared_v4i_p)(dst + i), 0, 0);
        }
        __builtin_amdgcn_s_wait_asynccnt(0);
#else
        for (int i = threadIdx.x; i < LDS_FLOATS / 4; i += THREADS) dst[i] = src[i];
#endif
    } else {
        stage_level_transpose(lds, emb0, LV0);
        stage_level_transpose(lds + N0, emb1, LV1);
    }
    __syncthreads();

    int base_pt = blockIdx.x * PPB;
#pragma unroll
    for (int k = 0; k < PPT; ++k) {
        int idx = base_pt + k * THREADS + threadIdx.x;
        if (idx >= n) break;

        // One-pass streaming reads: non-temporal (don't pollute L2)
        float x = __builtin_nontemporal_load(pts + idx * 3 + 0);
        float y = __builtin_nontemporal_load(pts + idx * 3 + 1);
        float z = __builtin_nontemporal_load(pts + idx * 3 + 2);

        float ax = fabsf(x), ay = fabsf(y), az = fabsf(z);
        bool is_x = (ax >= ay) && (ax >= az);
        bool is_y = !is_x && (ay >= az);

        int face;
        float ma;
        if (is_x)      { face = (x >= 0.0f) ? 0 : 1; ma = ax; }
        else if (is_y) { face = (y >= 0.0f) ? 2 : 3; ma = ay; }
        else           { face = (z >= 0.0f) ? 4 : 5; ma = az; }

        float sc = (face == 0) ? -z : (face == 1) ? z : (face == 5) ? -x : x;
        float tc = (face == 2) ?  z : (face == 3) ? -z : -y;

        float sm = fmaxf(ma, 1e-12f);
        float u = 0.5f * (sc / sm + 1.0f);
        float w = 0.5f * (tc / sm + 1.0f);

        float r[24];
        gather_chlast(lds,      face, u, w, LV0, r);
        gather_chlast(lds + N0, face, u, w, LV1, r + 6);
        if (USE_WS) {
            gather_chlast(tab2, face, u, w, LV2, r + 12);
            gather_chlast(tab3, face, u, w, LV3, r + 18);
        } else {
            gather_chmajor(tab2, face, u, w, LV2, r + 12);
            gather_chmajor(tab3, face, u, w, LV3, r + 18);
        }

        if (ma <= 0.0f) {
#pragma unroll
            for (int lvl = 0; lvl < 4; ++lvl)
#pragma unroll
                for (int c = 0; c < CCH; ++c) r[lvl * CCH + c] = fail[c];
        }

        // Write-once output stream: non-temporal b128 stores so the 192 MB
        // stream doesn't evict the L2-resident gather tables.
        vf4* o = (vf4*)(out + (size_t)idx * 24);
#pragma unroll
        for (int j = 0; j < 6; ++j) {
            vf4 v = {r[4 * j], r[4 * j + 1], r[4 * j + 2], r[4 * j + 3]};
            __builtin_nontemporal_store(v, o + j);
        }
    }
}

// ---------------- launch ----------------
extern "C" void kernel_launch(void* const* d_in, const int* in_sizes, int n_in,
                              void* d_out, int out_size, void* d_ws, size_t ws_size,
                              hipStream_t stream) {
    const float* pts  = (const float*)d_in[0];
    const float* e0   = (const float*)d_in[1];
    const float* e1   = (const float*)d_in[2];
    const float* e2   = (const float*)d_in[3];
    const float* e3   = (const float*)d_in[4];
    const float* fail = (const float*)d_in[5];
    float* out = (float*)d_out;

    int B = in_sizes[0] / 3;
    int grid = (B + PPB - 1) / PPB;

    bool use_ws = (d_ws != nullptr) && (ws_size >= (size_t)WS_FLOATS * sizeof(float));

    if (use_ws) {
        float* ws = (float*)d_ws;
        transpose_emb_kernel<<<1,    THREADS, 0, stream>>>(e0, ws + WS_OFF0, LV0);
        transpose_emb_kernel<<<6,    THREADS, 0, stream>>>(e1, ws + WS_OFF1, LV1);
        transpose_emb_kernel<<<96,   THREADS, 0, stream>>>(e2, ws + WS_OFF2, LV2);
        transpose_emb_kernel<<<1536, THREADS, 0, stream>>>(e3, ws + WS_OFF3, LV3);
        cubemap_encode_kernel<true><<<grid, THREADS, 0, stream>>>(
            pts, e0, e1, ws + WS_OFF2, ws + WS_OFF3, ws + WS_OFF0, fail, out, B);
    } else {
        cubemap_encode_kernel<false><<<grid, THREADS, 0, stream>>>(
            pts, e0, e1, e2, e3, nullptr, fail, out, B);
    }
}